// RankingLoss_10926396801141
// MI455X (gfx1250) — compile-verified
//
#include <hip/hip_runtime.h>
#include <hip/hip_bf16.h>

typedef __attribute__((ext_vector_type(2))) float v2f;
typedef __attribute__((ext_vector_type(8))) float v8f;

#define K1_THREADS 1024
#define K2_BLOCKS  512
#define K2_THREADS 256
#define CHUNK      16          // column tiles per work item (16*16 = 256 cols)
#define SENT 1.0e30f

// ---------------------------------------------------------------------------
// Kernel 1: stable label partition of pred into neg[] / pos[] via LDS scan.
// neg[] padded with +SENT to a multiple of 16 (strip height); pos[] padded
// with -SENT to a multiple of 256 (CHUNK*16) so kernel 2's inner loop has a
// compile-time trip count. Header {n_neg, n_pos, nnP16, npP256}.
// ---------------------------------------------------------------------------
__global__ __launch_bounds__(K1_THREADS)
void partition_kernel(const float* __restrict__ pred,
                      const int* __restrict__ target, int B,
                      float* __restrict__ negArr, float* __restrict__ posArr,
                      unsigned* __restrict__ hdr)
{
    __shared__ unsigned sScan[K1_THREADS];
    __shared__ unsigned sTotPos;
    const int tid   = threadIdx.x;
    const int items = (B + K1_THREADS - 1) / K1_THREADS;
    const int g0    = tid * items;
    const int g1    = (g0 + items < B) ? (g0 + items) : B;

    unsigned pcnt = 0;
    for (int g = g0; g < g1; ++g) pcnt += (target[g] != 0) ? 1u : 0u;
    sScan[tid] = pcnt;
    __syncthreads();
    // Hillis-Steele inclusive scan over per-thread pos counts
    for (int off = 1; off < K1_THREADS; off <<= 1) {
        unsigned v = (tid >= off) ? sScan[tid - off] : 0u;
        __syncthreads();
        sScan[tid] += v;
        __syncthreads();
    }
    const unsigned posExcl = sScan[tid] - pcnt;
    if (tid == K1_THREADS - 1) sTotPos = sScan[tid];
    __syncthreads();

    const unsigned nPos   = sTotPos;
    const unsigned nNeg   = (unsigned)B - nPos;
    const unsigned before = (unsigned)((g0 < B) ? g0 : B);
    unsigned pIdx = posExcl;
    unsigned nIdx = before - posExcl;
    for (int g = g0; g < g1; ++g) {
        float p = pred[g];
        if (target[g] != 0) posArr[pIdx++] = p;
        else                negArr[nIdx++] = p;
    }

    const unsigned nnP16  = (nNeg + 15u) & ~15u;
    const unsigned npP256 = (nPos + 255u) & ~255u;
    if (tid < 16) {
        unsigned kn = nNeg + (unsigned)tid;
        if (kn < nnP16) negArr[kn] = SENT;      // hinge & count contribute 0
    }
    if (tid < 256) {
        for (unsigned kp = nPos + (unsigned)tid; kp < npP256; kp += 256u)
            posArr[kp] = -SENT;                 // hinge & count contribute 0
    }
    if (tid == 0) { hdr[0] = nNeg; hdr[1] = nPos; hdr[2] = nnP16; hdr[3] = npP256; }
}

// ---------------------------------------------------------------------------
// Kernel 2: WMMA tile kernel. Work item = (16-row neg strip, chunk of exactly
// CHUNK=16 pos column tiles). A fragment loaded once per strip; inner loop is
// fully unrolled (compile-time trip count), division-free, EXEC-uniform:
// one b32 load + one v_wmma_f32_16x16x4_f32 per tile.
//   A row i = (n_i, 1, 0, 0);  B col j = (-1, q_j + 10, 0, 0)^T
//   s[i][j] = 10 + q_j - n_i
// Hinge: select-relu + fmac (VALU). Count: ballot+popc (SALU co-exec),
// wave-uniform accumulator. Deterministic fixed-order block reduction.
// ---------------------------------------------------------------------------
__global__ __launch_bounds__(K2_THREADS)
void pairloss_kernel(const float* __restrict__ negArr,
                     const float* __restrict__ posArr,
                     const unsigned* __restrict__ hdr,
                     float* __restrict__ blkLoss, unsigned* __restrict__ blkCnt)
{
    const unsigned nnP = hdr[2], npP = hdr[3];     // npP multiple of 256
    const unsigned tn      = nnP >> 4;             // row strips
    const unsigned nChunks = npP >> 8;             // 256-column chunks
    const unsigned totWork = tn * nChunks;

    const int tid        = threadIdx.x;
    const int lane       = tid & 31;
    const int lane16     = lane & 15;
    const bool lowHalf   = lane < 16;
    const int waveInBlk  = tid >> 5;
    const unsigned wpb   = K2_THREADS / 32;
    const unsigned gWave = blockIdx.x * wpb + (unsigned)waveInBlk;
    const unsigned nWav  = gridDim.x * wpb;

    float    lossAcc = 0.0f;
    unsigned cntAcc  = 0u;     // wave-uniform (ballot-based)

    for (unsigned w = gWave; w < totWork; w += nWav) {
        const unsigned si = w / nChunks;          // neg row strip (once/item)
        const unsigned ch = w - si * nChunks;     // column chunk
        const unsigned i0 = si << 4;
        const unsigned j0base = ch << 8;          // ch * 256 columns

        // A fragment: loaded once per strip; all lanes issue a valid load,
        // upper half-wave (K=2,3 columns) selected to zero.
        const float nval = negArr[i0 + (unsigned)lane16];
        v2f a;
        a.x = lowHalf ? nval : 0.0f;
        a.y = lowHalf ? 1.0f : 0.0f;

#pragma unroll
        for (unsigned k = 0; k < CHUNK; ++k) {
            const float qv = posArr[j0base + (k << 4) + (unsigned)lane16];
            v2f b;
            b.x = -1.0f;          // upper lanes inert (A K=2,3 == 0)
            b.y = qv + 10.0f;

            v8f c = {0.0f,0.0f,0.0f,0.0f,0.0f,0.0f,0.0f,0.0f};
            v8f d = __builtin_amdgcn_wmma_f32_16x16x4_f32(
                /*neg_a=*/false, a, /*neg_b=*/false, b,
                /*c_mod=*/(short)0, c, /*reuse_a=*/false, /*reuse_b=*/false);

#pragma unroll
            for (int r = 0; r < 8; ++r) {
                float s = d[r];                    // s = 10 + q_j - n_i
                float h = (s > 0.0f) ? s : 0.0f;   // cmp + cndmask
                lossAcc = fmaf(h, h, lossAcc);
                // count(q_j > n_i): v_cmp -> VCC, popc on scalar pipe
                cntAcc += (unsigned)__popcll(__ballot(s > 10.0f));
            }
        }
    }

    __shared__ float    sL[K2_THREADS];
    __shared__ unsigned sCw[K2_THREADS / 32];
    sL[tid] = lossAcc;
    if (lane == 0) sCw[waveInBlk] = cntAcc;
    __syncthreads();
    for (int s = K2_THREADS / 2; s > 0; s >>= 1) {
        if (tid < s) sL[tid] += sL[tid + s];
        __syncthreads();
    }
    if (tid == 0) {
        unsigned c = 0;
        for (int i = 0; i < K2_THREADS / 32; ++i) c += sCw[i];
        blkLoss[blockIdx.x] = sL[0];
        blkCnt[blockIdx.x]  = c;
    }
}

// ---------------------------------------------------------------------------
// Kernel 3: final deterministic reduction + division.
// loss_ref = 2*loss_sum, N = 2*nn*np  ->  out0 = loss_sum/(nn*np),
// out1 = cnt/(nn*np).
// ---------------------------------------------------------------------------
__global__ __launch_bounds__(256)
void finalize_kernel(const unsigned* __restrict__ hdr,
                     const float* __restrict__ blkLoss,
                     const unsigned* __restrict__ blkCnt,
                     int nblk, float* __restrict__ out)
{
    __shared__ float    sL[256];
    __shared__ unsigned sC[256];
    const int tid = threadIdx.x;
    float l = 0.0f; unsigned c = 0u;
    for (int i = tid; i < nblk; i += 256) { l += blkLoss[i]; c += blkCnt[i]; }
    sL[tid] = l; sC[tid] = c;
    __syncthreads();
    for (int s = 128; s > 0; s >>= 1) {
        if (tid < s) { sL[tid] += sL[tid + s]; sC[tid] += sC[tid + s]; }
        __syncthreads();
    }
    if (tid == 0) {
        float denom = (float)hdr[0] * (float)hdr[1];
        out[0] = sL[0] / denom;
        out[1] = (float)sC[0] / denom;
    }
}

extern "C" void kernel_launch(void* const* d_in, const int* in_sizes, int n_in,
                              void* d_out, int out_size, void* d_ws, size_t ws_size,
                              hipStream_t stream)
{
    const float* pred   = (const float*)d_in[0];
    const int*   target = (const int*)d_in[1];
    const int B = in_sizes[0];

    const int cap = ((B + 255) & ~255) + 256;   // per-array capacity (floats)
    char* ws = (char*)d_ws;
    float*    negArr  = (float*)ws;     ws += sizeof(float) * (size_t)cap;
    float*    posArr  = (float*)ws;     ws += sizeof(float) * (size_t)cap;
    unsigned* hdr     = (unsigned*)ws;  ws += sizeof(unsigned) * 4;
    float*    blkLoss = (float*)ws;     ws += sizeof(float) * K2_BLOCKS;
    unsigned* blkCnt  = (unsigned*)ws;  ws += sizeof(unsigned) * K2_BLOCKS;

    partition_kernel<<<1, K1_THREADS, 0, stream>>>(pred, target, B,
                                                   negArr, posArr, hdr);
    pairloss_kernel<<<K2_BLOCKS, K2_THREADS, 0, stream>>>(negArr, posArr, hdr,
                                                          blkLoss, blkCnt);
    finalize_kernel<<<1, 256, 0, stream>>>(hdr, blkLoss, blkCnt, K2_BLOCKS,
                                           (float*)d_out);
}